// MonarchConv_5849745457164
// MI455X (gfx1250) — compile-verified
//
#include <hip/hip_runtime.h>
#include <hip/hip_bf16.h>

// MonarchConv on gfx1250: FFT-conv with every DFT stage on V_WMMA_F32_16X16X4_F32.
// 2048 = 16 x 16 x 8 Cooley-Tukey; each stage = complex 16x16 @ 16x128 matmul,
// one workgroup (8 wave32) per (b,h) row, everything staged in LDS.
// Round-2 polish: split-plane (re/im/-im) LDS storage for the DFT matrices so
// A-fragments load directly as v2f (no repack movs / xors), and LDS-resident
// twiddle tables shared by the FFT and IFFT passes (4x fewer sincos).

typedef __attribute__((ext_vector_type(2))) float v2f_t;
typedef __attribute__((ext_vector_type(8))) float v8f_t;

#define PI_F 3.14159265358979f

__device__ __forceinline__ v8f_t wmma_f32(v2f_t a, v2f_t b, v8f_t c) {
  // (neg_a, A, neg_b, B, c_mod, C, reuse_a, reuse_b) -> v_wmma_f32_16x16x4_f32
  return __builtin_amdgcn_wmma_f32_16x16x4_f32(false, a, false, b, (short)0, c,
                                               false, false);
}

// ---- per-stage index maps (S = source buffer, T = dest buffer) --------------
// Stage 1: X[a][b]   = S[a*128+b]                (radix-16 over outer digit)
// Stage 2: col j=(o,d): X[c][j] = S[o*128+c*8+d] (16 independent 128-pt FFTs)
// Stage 3: block-diag(DFT8,DFT8); col j packs two (o,p) groups t=2j+s.

template <int STAGE>
__device__ __forceinline__ float2 ldB(const float2* S, int k, int col) {
  if constexpr (STAGE == 1) {
    return S[k * 128 + col];
  } else if constexpr (STAGE == 2) {
    int o = col >> 3, d = col & 7;
    return S[o * 128 + k * 8 + d];
  } else {
    int s = k >> 3, d = k & 7;
    int t = 2 * col + s;
    int o = t >> 4, p = t & 15;
    return S[o * 128 + p * 8 + d];
  }
}

template <int STAGE>
__device__ __forceinline__ void stC(float2* T, int row, int col, float2 v) {
  if constexpr (STAGE == 1) {
    T[row * 128 + col] = v;                 // layout [o][b]
  } else if constexpr (STAGE == 2) {
    int o = col >> 3, d = col & 7;
    T[o * 128 + row * 8 + d] = v;           // layout [o][p][d]
  } else {
    int s = row >> 3, q = row & 7;
    int t = 2 * col + s;
    int o = t >> 4, p = t & 15;
    T[q * 256 + p * 16 + o] = v;            // natural-order spectrum
  }
}

// Twiddle from LDS tables: tw1[row*128+col] = exp(-2i*pi*row*col/2048),
// tw2[row*8+d] = exp(-2i*pi*row*d/128).
template <int STAGE>
__device__ __forceinline__ float2 twd(const float2* __restrict__ tw1,
                                      const float2* __restrict__ tw2, int row,
                                      int col, float2 z) {
  if constexpr (STAGE == 3) return z;
  float2 t;
  if constexpr (STAGE == 1)
    t = tw1[row * 128 + col];
  else
    t = tw2[row * 8 + (col & 7)];
  return make_float2(z.x * t.x - z.y * t.y, z.x * t.y + z.y * t.x);
}

// One radix stage: wave `wave` computes output tile columns [16w,16w+16).
// Complex D = M*X via 4 real WMMA accumulations per K-chunk (K=16 -> 4 chunks).
// A3 = 3 planes of 256 floats: [0]=re, [256]=im, [512]=-im, row-major 16x16.
template <int STAGE>
__device__ __forceinline__ void run_stage(const float* __restrict__ A3,
                                          const float2* __restrict__ S,
                                          float2* __restrict__ T,
                                          const float2* __restrict__ tw1,
                                          const float2* __restrict__ tw2,
                                          int lane, int wave) {
  const int col = wave * 16 + (lane & 15);
  const int hi = lane >> 4;  // lanes 16-31 carry K+2 / M+8 halves
  const int rowA = lane & 15;
  v8f_t cre = {};
  v8f_t cim = {};
#pragma unroll
  for (int c4 = 0; c4 < 4; ++c4) {
    const int k0 = 4 * c4 + 2 * hi;        // even -> 8B-aligned plane loads
    const int ai = rowA * 16 + k0;
    v2f_t ar = *(const v2f_t*)(A3 + ai);          // {A[M][k0],A[M][k0+1]}.re
    v2f_t aim = *(const v2f_t*)(A3 + 256 + ai);   // .im
    v2f_t nai = *(const v2f_t*)(A3 + 512 + ai);   // -.im (f32 WMMA has no NEG)
    float2 b0 = ldB<STAGE>(S, k0, col);
    float2 b1 = ldB<STAGE>(S, k0 + 1, col);
    v2f_t br = {b0.x, b1.x};
    v2f_t bi = {b0.y, b1.y};
    cre = wmma_f32(ar, br, cre);
    cre = wmma_f32(nai, bi, cre);
    cim = wmma_f32(ar, bi, cim);
    cim = wmma_f32(aim, br, cim);
  }
#pragma unroll
  for (int v = 0; v < 8; ++v) {
    int row = v + 8 * hi;  // C/D layout: VGPR v -> M=v (lanes<16) / v+8
    float2 z = make_float2(cre[v], cim[v]);
    stC<STAGE>(T, row, col, twd<STAGE>(tw1, tw2, row, col, z));
  }
}

__device__ __forceinline__ void fft2048(float2* b0, float2* b1,
                                        const float* A16, const float* A8,
                                        const float2* tw1, const float2* tw2,
                                        int lane, int wave) {
  run_stage<1>(A16, b0, b1, tw1, tw2, lane, wave);
  __syncthreads();
  run_stage<2>(A16, b1, b0, tw1, tw2, lane, wave);
  __syncthreads();
  run_stage<3>(A8, b0, b1, tw1, tw2, lane, wave);
  __syncthreads();
}

// Load per-h DFT matrices into split-plane LDS and build twiddle tables.
__device__ __forceinline__ void load_consts(const float2* __restrict__ mat16,
                                            const float2* __restrict__ mat8,
                                            float* A16, float* A8, float2* tw1,
                                            float2* tw2, int h, int tid) {
  float2 m = mat16[h * 256 + tid];
  A16[tid] = m.x;
  A16[256 + tid] = m.y;
  A16[512 + tid] = -m.y;
  int e = tid >> 4, f = tid & 15;
  float2 v = make_float2(0.f, 0.f);
  if ((e >> 3) == (f >> 3)) v = mat8[h * 64 + (e & 7) * 8 + (f & 7)];
  A8[tid] = v.x;             // block-diag(DFT8, DFT8)
  A8[256 + tid] = v.y;
  A8[512 + tid] = -v.y;
  for (int i = tid; i < 2048; i += 256) {
    int r = i >> 7, c = i & 127;
    float s, co;
    __sincosf((-2.0f * PI_F / 2048.0f) * (float)(r * c), &s, &co);
    tw1[i] = make_float2(co, s);
  }
  if (tid < 128) {
    int r = tid >> 3, d = tid & 7;
    float s, co;
    __sincosf((-2.0f * PI_F / 128.0f) * (float)(r * d), &s, &co);
    tw2[tid] = make_float2(co, s);
  }
}

__global__ __launch_bounds__(256) void monarch_kfft(
    const float* __restrict__ kin, const float2* __restrict__ mat16,
    const float2* __restrict__ mat8, float2* __restrict__ kf) {
  __shared__ float2 sB0[2048];
  __shared__ float2 sB1[2048];
  __shared__ float sA16[3 * 256];
  __shared__ float sA8[3 * 256];
  __shared__ float2 sTw1[2048];
  __shared__ float2 sTw2[128];
  const int h = blockIdx.x;
  const int tid = threadIdx.x;
  const int lane = tid & 31, wave = tid >> 5;
  load_consts(mat16, mat8, sA16, sA8, sTw1, sTw2, h, tid);
  for (int i = tid; i < 2048; i += 256) {
    float re = (i < 1024) ? kin[h * 1024 + i] : 0.f;
    sB0[i] = make_float2(re, 0.f);
  }
  __syncthreads();
  fft2048(sB0, sB1, sA16, sA8, sTw1, sTw2, lane, wave);
  for (int i = tid; i < 2048; i += 256) kf[h * 2048 + i] = sB1[i];
}

__global__ __launch_bounds__(256) void monarch_main(
    const float* __restrict__ u, const float* __restrict__ D,
    const float2* __restrict__ mat16, const float2* __restrict__ mat8,
    const float2* __restrict__ kf, float* __restrict__ y) {
  __shared__ float2 sB0[2048];
  __shared__ float2 sB1[2048];
  __shared__ float sA16[3 * 256];
  __shared__ float sA8[3 * 256];
  __shared__ float2 sTw1[2048];
  __shared__ float2 sTw2[128];
  const int h = blockIdx.x;
  const int b = blockIdx.y;
  const int tid = threadIdx.x;
  const int lane = tid & 31, wave = tid >> 5;
  const long base = ((long)b * 512 + h) * 1024;
  load_consts(mat16, mat8, sA16, sA8, sTw1, sTw2, h, tid);
  for (int i = tid; i < 2048; i += 256) {
    float re = (i < 1024) ? u[base + i] : 0.f;
    sB0[i] = make_float2(re, 0.f);
  }
  // Warm GL2 with this row's filter spectrum (global_prefetch_b8).
  __builtin_prefetch(&kf[h * 2048 + tid * 8], 0, 0);
  __syncthreads();

  fft2048(sB0, sB1, sA16, sA8, sTw1, sTw2, lane, wave);  // U = FFT(pad(u))

  // P = U * Kf; feed conj(P) to the same forward pipeline -> IFFT via conj.
  for (int i = tid; i < 2048; i += 256) {
    float2 U = sB1[i];
    float2 K = kf[h * 2048 + i];
    sB0[i] = make_float2(U.x * K.x - U.y * K.y, -(U.x * K.y + U.y * K.x));
  }
  __syncthreads();

  fft2048(sB0, sB1, sA16, sA8, sTw1, sTw2, lane, wave);  // conj(IFFT)*N

  const float d = D[h];
  const float invN = 1.0f / 2048.0f;
  for (int i = tid; i < 1024; i += 256)
    y[base + i] = sB1[i].x * invN + u[base + i] * d;
}

extern "C" void kernel_launch(void* const* d_in, const int* in_sizes, int n_in,
                              void* d_out, int out_size, void* d_ws,
                              size_t ws_size, hipStream_t stream) {
  const float* u = (const float*)d_in[0];
  const float* k = (const float*)d_in[1];
  const float* D = (const float*)d_in[2];
  // d_in[3]=mat_2, d_in[4]=mat_4 unused: 2048 factors as 16*16*8.
  const float2* mat8 = (const float2*)d_in[5];   // (H,8,8,2) -> float2
  const float2* mat16 = (const float2*)d_in[6];  // (H,16,16,2) -> float2
  float* y = (float*)d_out;
  float2* kf = (float2*)d_ws;  // H*2048 complex64 = 8 MB filter spectra
  const int H = 512, B = 16;
  monarch_kfft<<<dim3(H), dim3(256), 0, stream>>>(k, mat16, mat8, kf);
  monarch_main<<<dim3(H, B), dim3(256), 0, stream>>>(u, D, mat16, mat8, kf, y);
}